// AvgPool2d_20203526160806
// MI455X (gfx1250) — compile-verified
//
#include <hip/hip_runtime.h>
#include <stdint.h>

// AvgPool 64x64, stride 1, edge-replicate pad back to 256x256 over
// (16,64,256,256) fp32. Bandwidth-bound: 512MB traffic -> ~22us @ 23.3TB/s.
//
// Design (one 256-thread workgroup = 8 wave32 per (n,c) image):
//  - whole 256x256 fp32 image staged in LDS (CDNA5: 320KB/WGP; we use
//    256*289*4 = 295,936B; stride 289 -> 33 mod 64 (odd) so both row-wise
//    and column-wise lane access are bank-conflict-free; 32-col margin)
//  - Stage A: CDNA5 async global->LDS B128 DMA (ASYNCcnt-tracked, bypasses
//    VGPRs), induction-variable addressing: +4096B global / +4624B LDS per
//    step, 32 lanes fetch 512B contiguous global per wave issue.
//  - Stage B: horizontal 64-wide running sum, written in place shifted left
//    by 32 (write at col p destroys only x[p-32], needed only by past
//    steps); the margin becomes the left replicate-pad.
//  - Stage C: vertical 64-tall running sum, *1/4096 (exact), coalesced
//    non-temporal stores straight to global output.

#define IMG_H   256
#define IMG_W   256
#define KWIN    64
#define LSTRIDE 289   // 256 data cols + 32 margin + 1 skew (odd mod 64)

__global__ __launch_bounds__(256)
void avgpool64_box_kernel(const float* __restrict__ x, float* __restrict__ out) {
    extern __shared__ float tile[];              // [256][LSTRIDE], LDS offset 0
    const int t   = threadIdx.x;                 // 0..255
    const int img = blockIdx.x;                  // n*64 + c
    const float* src = x   + (size_t)img * (IMG_H * IMG_W);
    float*       dst = out + (size_t)img * (IMG_H * IMG_W);

    // ---------------- Stage A: async DMA image -> LDS ----------------
    // chunk id c = k*256 + t (16B each): row = 4k + (t>>6), j = t&63.
    // Global:  src + t*16 + k*4096   (lanes: contiguous 512B per issue)
    // LDS:     (row*LSTRIDE + 32)*4 + j*16  ->  base + k*4624
    {
        unsigned lds_byte = (unsigned)(((t >> 6) * LSTRIDE + 32) * 4 + (t & 63) * 16);
        unsigned long long gaddr = (unsigned long long)(uintptr_t)src
                                 + (unsigned long long)(t * 16);
        #pragma unroll 8
        for (int k = 0; k < 64; ++k) {
            asm volatile("global_load_async_to_lds_b128 %0, %1, off"
                         :: "v"(lds_byte), "v"(gaddr) : "memory");
            lds_byte += 4u * LSTRIDE * 4u;   // 4 rows = 4624 bytes
            gaddr    += 4096ull;             // 256 chunks * 16B
        }
    }
    asm volatile("s_wait_asynccnt 0x0" ::: "memory");
    __syncthreads();

    // ---------------- Stage B: horizontal running window (row t) -----
    // x[j] at col 32+j. out[p] = v[clamp(p-31,0,192)], v[i] = sum x[i..i+63].
    // out[p] written at col p: destroys only x[p-32] (past-only dependence).
    {
        float* row = tile + t * LSTRIDE;
        float s = 0.0f;
        #pragma unroll
        for (int j = 0; j < KWIN; ++j) s += row[32 + j];     // v[0]
        #pragma unroll
        for (int p = 0; p < 32; ++p)   row[p] = s;           // left pad = v[0]
        #pragma unroll 8
        for (int p = 32; p <= 223; ++p) {                    // v[p-31]
            s += row[p + 64] - row[p];                       // +x[p+32] -x[p-32]
            row[p] = s;
        }
        #pragma unroll
        for (int p = 224; p < 256; ++p) row[p] = s;          // right pad = v[192]
    }
    __syncthreads();

    // ---------------- Stage C: vertical running window (col t) -------
    {
        const float inv = 1.0f / (float)(KWIN * KWIN);       // 1/4096 exact
        float s = 0.0f;
        #pragma unroll
        for (int r = 0; r < KWIN; ++r) s += tile[r * LSTRIDE + t];   // v[0]
        float v = s * inv;
        #pragma unroll
        for (int p = 0; p < 32; ++p)
            __builtin_nontemporal_store(v, dst + p * IMG_W + t);
        #pragma unroll 8
        for (int p = 32; p <= 223; ++p) {
            s += tile[(p + 32) * LSTRIDE + t] - tile[(p - 32) * LSTRIDE + t];
            __builtin_nontemporal_store(s * inv, dst + p * IMG_W + t);
        }
        v = s * inv;
        #pragma unroll
        for (int p = 224; p < 256; ++p)
            __builtin_nontemporal_store(v, dst + p * IMG_W + t);
    }
}

extern "C" void kernel_launch(void* const* d_in, const int* in_sizes, int n_in,
                              void* d_out, int out_size, void* d_ws, size_t ws_size,
                              hipStream_t stream) {
    (void)n_in; (void)d_ws; (void)ws_size; (void)out_size;
    const float* x   = (const float*)d_in[0];
    float*       out = (float*)d_out;
    const int nimg = in_sizes[0] / (IMG_H * IMG_W);          // 16*64 = 1024
    const size_t lds_bytes = (size_t)IMG_H * LSTRIDE * sizeof(float); // 295,936B
    avgpool64_box_kernel<<<nimg, 256, lds_bytes, stream>>>(x, out);
}